// XYLoss_63754494541943
// MI455X (gfx1250) — compile-verified
//
#include <hip/hip_runtime.h>

typedef __attribute__((ext_vector_type(2))) float v2f;
typedef __attribute__((ext_vector_type(4))) float v4f;
typedef __attribute__((ext_vector_type(8))) float v8f;

#define HW_   (1024 * 1024)
#define B_    16
#define NJ_   (B_ * HW_)           // 16,777,216 spatial elements
#define NJ4_  (NJ_ / 4)            // 4,194,304 float4 groups
#define NBLK  1024
#define NTHR  256                  // 8 waves of 32

// ---- fast transcendentals straight onto the trans pipe (v_exp_f32 / v_log_f32 / v_rcp_f32)
__device__ __forceinline__ float fexp(float x) {
  return __builtin_amdgcn_exp2f(x * 1.4426950408889634f);
}
__device__ __forceinline__ float flog(float x) {
  return __builtin_amdgcn_logf(x) * 0.6931471805599453f;
}
__device__ __forceinline__ float sigmoidf_fast(float x) {
  // 1 / (1 + exp(-x))
  return __builtin_amdgcn_rcpf(1.0f + fexp(-x));
}

// ---- 32-lane sum using V_WMMA_F32_16X16X4_F32
// A: VGPR0 = per-lane partial, VGPR1 = 0  => A[m][0]=x[m], A[m][2]=x[m+16]
// B: all-ones 4x16
// D[m][n] = x[m] + x[m+16] = s[m], replicated across n.
// C/D layout: lane L<16 holds D[r][L] in VGPR r; lane L>=16 holds D[r+8][L-16].
// Sum of a lane's 8 D-VGPRs: lanes 0..15 -> sum(s[0..7]); lanes 16..31 -> sum(s[8..15]).
// One xor-16 shuffle completes the reduction; result valid in all lanes.
__device__ __forceinline__ float wave_sum32(float x) {
  v2f a; a[0] = x;    a[1] = 0.0f;
  v2f b; b[0] = 1.0f; b[1] = 1.0f;
  v8f c = {};
  v8f d = __builtin_amdgcn_wmma_f32_16x16x4_f32(
      /*neg_a=*/false, a, /*neg_b=*/false, b,
      /*c_mod=*/(short)0, c, /*reuse_a=*/false, /*reuse_b=*/false);
  float r = ((d[0] + d[1]) + (d[2] + d[3])) + ((d[4] + d[5]) + (d[6] + d[7]));
  r += __shfl_xor(r, 16, 32);
  return r;
}

// Streamed-once data (384 MB total, 2x the 192 MB L2): non-temporal loads so
// the one-pass stream doesn't rinse L2. Native clang vector type required by
// __builtin_nontemporal_load.
__device__ __forceinline__ v4f ldnt4(const float* p) {
  return __builtin_nontemporal_load((const v4f*)p);
}

// ---- Pass 1: stream everything once, produce per-block partial (S1, S2)
__global__ void __launch_bounds__(NTHR)
xy_loss_pass1(const float* __restrict__ mask,
              const float* __restrict__ scale,
              const float* __restrict__ pred,
              const float* __restrict__ tru,
              float* __restrict__ partials) {
  const int tid   = blockIdx.x * NTHR + threadIdx.x;
  const int nthr  = NBLK * NTHR;          // 262,144 -> exactly 16 iterations each
  float acc1 = 0.0f;                      // Σ per_elem
  float acc2 = 0.0f;                      // Σ mask*scale

  for (int v = tid; v < NJ4_; v += nthr) {
    const int j  = v << 2;                // element index in [0, B*HW)
    const int b  = j >> 20;               // j / HW
    const int hw = j & (HW_ - 1);         // j % HW
    const int i0 = b * (2 * HW_) + hw;    // channel-0 flat index (fits in int)

    const v4f x0 = ldnt4(pred  + i0);
    const v4f x1 = ldnt4(pred  + i0 + HW_);
    const v4f t0 = ldnt4(tru   + i0);
    const v4f t1 = ldnt4(tru   + i0 + HW_);
    const v4f mm = ldnt4(mask  + j);
    const v4f ss = ldnt4(scale + j);

    acc2 += (mm[0] * ss[0] + mm[1] * ss[1]) + (mm[2] * ss[2] + mm[3] * ss[3]);

#pragma unroll
    for (int k = 0; k < 4; ++k) {
      const float p0 = sigmoidf_fast(x0[k]);
      const float p1 = sigmoidf_fast(x1[k]);
      const float mx = fmaxf(p0, p1);
      const float mn = fminf(p0, p1);
      const float lse = mx + flog(1.0f + fexp(mn - mx));
      acc1 += (t0[k] + t1[k]) * lse - (t0[k] * p0 + t1[k] * p1);
    }
  }

  // intra-wave reduction via WMMA (EXEC is all-ones here: uniform control flow)
  const float w1 = wave_sum32(acc1);
  const float w2 = wave_sum32(acc2);

  __shared__ float red1[NTHR / 32];
  __shared__ float red2[NTHR / 32];
  const int lane = threadIdx.x & 31;
  const int wid  = threadIdx.x >> 5;
  if (lane == 0) { red1[wid] = w1; red2[wid] = w2; }
  __syncthreads();

  if (threadIdx.x == 0) {
    float s1 = 0.0f, s2 = 0.0f;
#pragma unroll
    for (int i = 0; i < NTHR / 32; ++i) { s1 += red1[i]; s2 += red2[i]; }
    partials[2 * blockIdx.x + 0] = s1;
    partials[2 * blockIdx.x + 1] = s2;
  }
}

// ---- Pass 2: one wave, fixed partition + fixed reduction tree => deterministic.
// Each lane sums a contiguous chunk of 32 block-partials in order, then the
// same WMMA wave reduction combines the 32 lane partials.
__global__ void __launch_bounds__(32)
xy_loss_pass2(const float* __restrict__ partials,
              float* __restrict__ out) {
  const int t = threadIdx.x;              // 0..31, single wave, EXEC all-ones
  float s1 = 0.0f, s2 = 0.0f;
#pragma unroll 4
  for (int i = 0; i < NBLK / 32; ++i) {
    const int idx = t * (NBLK / 32) + i;  // fixed per-lane chunk, fixed order
    s1 += partials[2 * idx + 0];
    s2 += partials[2 * idx + 1];
  }
  const float S1 = wave_sum32(s1);
  const float S2 = wave_sum32(s2);
  if (t == 0) out[0] = (S1 / (float)NJ_) * S2;
}

extern "C" void kernel_launch(void* const* d_in, const int* in_sizes, int n_in,
                              void* d_out, int out_size, void* d_ws, size_t ws_size,
                              hipStream_t stream) {
  const float* mask  = (const float*)d_in[0];  // object_mask    [16,1,1024,1024]
  const float* scale = (const float*)d_in[1];  // box_loss_scale [16,1,1024,1024]
  const float* pred  = (const float*)d_in[2];  // predict_xy     [16,2,1024,1024]
  const float* tru   = (const float*)d_in[3];  // true_xy        [16,2,1024,1024]
  float* partials = (float*)d_ws;              // 2 * NBLK floats (8 KB)
  float* out = (float*)d_out;

  xy_loss_pass1<<<NBLK, NTHR, 0, stream>>>(mask, scale, pred, tru, partials);
  xy_loss_pass2<<<1, 32, 0, stream>>>(partials, out);
}